// HeadAttetion_73727408603351
// MI455X (gfx1250) — compile-verified
//
#include <hip/hip_runtime.h>
#include <math.h>

typedef float v2f __attribute__((ext_vector_type(2)));
typedef float v8f __attribute__((ext_vector_type(8)));

#define EMB   1024
#define KD    64
#define BATCH 8
#define SEQ   2048
#define ROWS  (BATCH * SEQ)   // 16384

// ---------------------------------------------------------------------------
// Kernel A: WP[mat][i][d] = sum_j W[mat][i][j] * P[mat][j][d]   (3 x 1024 x 64)
// ---------------------------------------------------------------------------
__global__ __launch_bounds__(256) void fuse_wp(
    const float* __restrict__ Wq, const float* __restrict__ Wk, const float* __restrict__ Wv,
    const float* __restrict__ Pq, const float* __restrict__ Pk, const float* __restrict__ Pv,
    float* __restrict__ WP) {
  int idx = blockIdx.x * 256 + threadIdx.x;      // 0 .. 3*1024*64-1
  int mat = idx >> 16;
  int rem = idx & 65535;
  int i = rem >> 6;
  int d = rem & 63;
  const float* W = (mat == 0) ? Wq : (mat == 1) ? Wk : Wv;
  const float* P = (mat == 0) ? Pq : (mat == 1) ? Pk : Pv;
  const float* wrow = W + i * EMB;
  float acc = 0.0f;
#pragma unroll 8
  for (int j = 0; j < EMB; ++j) acc = fmaf(wrow[j], P[j * KD + d], acc);
  WP[idx] = acc;
}

// ---------------------------------------------------------------------------
// Kernel A2: bP[mat][d] = sum_j b[mat][j] * P[mat][j][d]
// ---------------------------------------------------------------------------
__global__ __launch_bounds__(192) void fuse_bp(
    const float* __restrict__ bq, const float* __restrict__ bk, const float* __restrict__ bv,
    const float* __restrict__ Pq, const float* __restrict__ Pk, const float* __restrict__ Pv,
    float* __restrict__ bP) {
  int idx = threadIdx.x;                          // 0..191
  int mat = idx >> 6;
  int d = idx & 63;
  const float* b = (mat == 0) ? bq : (mat == 1) ? bk : bv;
  const float* P = (mat == 0) ? Pq : (mat == 1) ? Pk : Pv;
  float acc = 0.0f;
#pragma unroll 8
  for (int j = 0; j < EMB; ++j) acc = fmaf(b[j], P[j * KD + d], acc);
  bP[idx] = acc;
}

// ---------------------------------------------------------------------------
// Kernel B: qkv[mat][row][d] = x[row][:] @ WP[mat][:, d] + bP[mat][d]
// Grid: 1024 row-tiles of 16. Block: 384 threads = 12 waves.
// Wave w -> mat = w>>2, n-tile = (w&3)*16. K loop via v_wmma_f32_16x16x4_f32.
// ---------------------------------------------------------------------------
__global__ __launch_bounds__(384) void qkv_gemm(
    const float* __restrict__ x, const float* __restrict__ WP,
    const float* __restrict__ bP, float* __restrict__ qkv) {
  __shared__ float xs[16 * EMB];                  // 64 KB x-tile

  const int tid  = threadIdx.x;
  const int row0 = blockIdx.x * 16;

  // Cooperative float4 load of the 16x1024 x tile.
  const float4* xg  = (const float4*)(x + (size_t)row0 * EMB);
  float4*       xs4 = (float4*)xs;
#pragma unroll
  for (int i = tid; i < (16 * EMB) / 4; i += 384) xs4[i] = xg[i];
  __syncthreads();

  const int wave = tid >> 5;
  const int lane = tid & 31;
  const int mat  = wave >> 2;
  const int n0   = (wave & 3) * 16;
  const int m    = lane & 15;           // A row (M)
  const int koff = (lane >> 4) * 2;     // lanes 16-31 hold K=2,3
  const int n    = lane & 15;           // B / C column (N)

  const float* Wm   = WP + (size_t)mat * (EMB * KD);
  const float* xrow = xs + m * EMB + koff;
  const float* bcol = Wm + (size_t)koff * KD + n0 + n;

  v8f acc = {};
  for (int k0 = 0; k0 < EMB; k0 += 4) {
    v2f a;  a.x  = xrow[k0];            a.y  = xrow[k0 + 1];
    v2f bb; bb.x = bcol[(size_t)k0 * KD]; bb.y = bcol[(size_t)k0 * KD + KD];
    acc = __builtin_amdgcn_wmma_f32_16x16x4_f32(false, a, false, bb,
                                                (short)0, acc, false, false);
  }

  const float bias = bP[mat * KD + n0 + n];
  float* out = qkv + (size_t)mat * (ROWS * KD)
                   + (size_t)(row0 + (lane >> 4) * 8) * KD + n0 + n;
#pragma unroll
  for (int v = 0; v < 8; ++v) out[(size_t)v * KD] = acc[v] + bias;
}

// ---------------------------------------------------------------------------
// Kernel C: Mkv[b][d1][d2] = (1/8) * sum_t kw[b][t][d1] * vw[b][t][d2]
// Grid: 8 batches. Block: 512 threads = 16 waves (4x4 tiles of the 64x64 out).
// ---------------------------------------------------------------------------
__global__ __launch_bounds__(512) void kv_outer(
    const float* __restrict__ qkv, float* __restrict__ Mkv) {
  const int b = blockIdx.x;
  const float* kwb = qkv + (size_t)1 * (ROWS * KD) + (size_t)b * SEQ * KD;
  const float* vwb = qkv + (size_t)2 * (ROWS * KD) + (size_t)b * SEQ * KD;

  const int wave = threadIdx.x >> 5;
  const int lane = threadIdx.x & 31;
  const int d1_0 = (wave >> 2) * 16;
  const int d2_0 = (wave & 3) * 16;
  const int m    = lane & 15;
  const int koff = (lane >> 4) * 2;
  const int n    = lane & 15;

  const float* ka = kwb + (size_t)koff * KD + d1_0 + m;   // A[m][k] = kw[k][d1_0+m]
  const float* vb = vwb + (size_t)koff * KD + d2_0 + n;   // B[k][n] = vw[k][d2_0+n]

  v8f acc = {};
  for (int t0 = 0; t0 < SEQ; t0 += 4) {
    v2f a;  a.x  = ka[(size_t)t0 * KD]; a.y  = ka[(size_t)t0 * KD + KD];
    v2f bb; bb.x = vb[(size_t)t0 * KD]; bb.y = vb[(size_t)t0 * KD + KD];
    acc = __builtin_amdgcn_wmma_f32_16x16x4_f32(false, a, false, bb,
                                                (short)0, acc, false, false);
  }

  const float inv_scale = 0.125f;        // 1/sqrt(KD)
  float* out = Mkv + (size_t)b * KD * KD
                   + (size_t)(d1_0 + (lane >> 4) * 8) * KD + d2_0 + n;
#pragma unroll
  for (int v = 0; v < 8; ++v) out[(size_t)v * KD] = acc[v] * inv_scale;
}

// ---------------------------------------------------------------------------
// Kernel D: z = qw @ Mkv[b]; soft = softmax(z, last dim 64); out = soft@Wo + bo
// Block: 256 threads = 4 rows x 64 cols.
// ---------------------------------------------------------------------------
__global__ __launch_bounds__(256) void attn_out(
    const float* __restrict__ qkv, const float* __restrict__ Mkv,
    const float* __restrict__ Wo, const float* __restrict__ bo,
    float* __restrict__ out) {
  __shared__ float qs[4][64];
  __shared__ float es[4][64];

  const int tid = threadIdx.x;
  const int lr  = tid >> 6;
  const int col = tid & 63;
  const int r   = blockIdx.x * 4 + lr;
  const int b   = r >> 11;               // / SEQ

  qs[lr][col] = qkv[(size_t)r * KD + col];   // qw rows
  __syncthreads();

  const float* M = Mkv + (size_t)b * KD * KD;
  float z = 0.0f;
#pragma unroll 8
  for (int d = 0; d < KD; ++d) z = fmaf(qs[lr][d], M[d * KD + col], z);

  es[lr][col] = z;
  __syncthreads();

  float mx = -INFINITY;
#pragma unroll 8
  for (int d = 0; d < KD; ++d) mx = fmaxf(mx, es[lr][d]);
  __syncthreads();

  const float e = expf(z - mx);
  es[lr][col] = e;
  __syncthreads();

  float s = 0.0f, acc = 0.0f;
#pragma unroll 8
  for (int d = 0; d < KD; ++d) {
    const float ed = es[lr][d];
    s   += ed;
    acc  = fmaf(ed, Wo[d * KD + col], acc);
  }
  out[(size_t)r * KD + col] = acc / s + bo[col];
}

// ---------------------------------------------------------------------------
extern "C" void kernel_launch(void* const* d_in, const int* in_sizes, int n_in,
                              void* d_out, int out_size, void* d_ws, size_t ws_size,
                              hipStream_t stream) {
  const float* x  = (const float*)d_in[0];
  const float* Wq = (const float*)d_in[1];
  const float* bq = (const float*)d_in[2];
  const float* Wk = (const float*)d_in[3];
  const float* bk = (const float*)d_in[4];
  const float* Wv = (const float*)d_in[5];
  const float* bv = (const float*)d_in[6];
  const float* Pq = (const float*)d_in[7];
  const float* Pk = (const float*)d_in[8];
  const float* Pv = (const float*)d_in[9];
  const float* Wo = (const float*)d_in[10];
  const float* bo = (const float*)d_in[11];
  float* out = (float*)d_out;

  float* ws  = (float*)d_ws;
  float* WP  = ws;                         // 3*1024*64   = 196608 floats
  float* bP  = ws + 196608;                // 3*64 (padded to 192)
  float* qkv = ws + 196800;                // 3*16384*64  = 3145728 floats
  float* Mkv = ws + 196800 + 3145728;      // 8*64*64     = 32768 floats

  fuse_wp <<<768, 256, 0, stream>>>(Wq, Wk, Wv, Pq, Pk, Pv, WP);
  fuse_bp <<<1, 192, 0, stream>>>(bq, bk, bv, Pq, Pk, Pv, bP);
  qkv_gemm<<<ROWS / 16, 384, 0, stream>>>(x, WP, bP, qkv);
  kv_outer<<<BATCH, 512, 0, stream>>>(qkv, Mkv);
  attn_out<<<ROWS / 4, 256, 0, stream>>>(qkv, Mkv, Wo, bo, out);
}